// GatedLayer_33552284516386
// MI455X (gfx1250) — compile-verified
//
#include <hip/hip_runtime.h>
#include <math.h>

// Problem constants (match reference)
#define NN 50000
#define DD 32        // neighbors per node == wave32 width (1 wave <-> 1 node)
#define FF 128       // features per node  == 32 lanes * float4
#define CC 16        // classes
#define LN_EPS 1e-5f
#define CLAMP_MIN 1e-5f

#define WAVES_PER_BLOCK 8
#define BLOCK (WAVES_PER_BLOCK * 32)

// ---------- wave32 helpers (gfx1250 is wave32-only) ----------
__device__ __forceinline__ unsigned int ballot32(bool p) {
#if __has_builtin(__builtin_amdgcn_ballot_w32)
  return __builtin_amdgcn_ballot_w32(p);
#else
  return (unsigned int)__ballot(p);
#endif
}

__device__ __forceinline__ int bcast_lane(int v, int lane) {
#if __has_builtin(__builtin_amdgcn_readlane)
  return __builtin_amdgcn_readlane(v, lane);   // v_readlane -> SGPR, enables saddr loads
#else
  return __shfl(v, lane, 32);
#endif
}

// First-occurrence argmax over 16 contiguous floats (matches jnp.argmax)
__device__ __forceinline__ int argmax16(const float4* __restrict__ lp, int row) {
  float4 a = lp[row * 4 + 0];
  float4 b = lp[row * 4 + 1];
  float4 c = lp[row * 4 + 2];
  float4 d = lp[row * 4 + 3];
  float v[16] = {a.x, a.y, a.z, a.w, b.x, b.y, b.z, b.w,
                 c.x, c.y, c.z, c.w, d.x, d.y, d.z, d.w};
  int best = 0;
  float bv = v[0];
#pragma unroll
  for (int k = 1; k < 16; ++k) {
    if (v[k] > bv) { bv = v[k]; best = k; }   // strict > keeps first max
  }
  return best;
}

// ---------- kernel 0: zero the reduction accumulators ----------
__global__ void k_init(float* __restrict__ stats) {
  if (threadIdx.x < 4) stats[threadIdx.x] = 0.0f;
}

// ---------- kernel 1: per-node f1/f2 + global sum/sumsq reduction ----------
__global__ __launch_bounds__(BLOCK) void k_stats(
    const float* __restrict__ logits, const int* __restrict__ nbr,
    float* __restrict__ f1o, float* __restrict__ f2o, float* __restrict__ stats) {
  const int wave = threadIdx.x >> 5;
  const int lane = threadIdx.x & 31;
  const int i = blockIdx.x * WAVES_PER_BLOCK + wave;
  __shared__ float red[WAVES_PER_BLOCK][4];

  float wf1 = 0.0f, wf2 = 0.0f;
  if (i < NN) {
    const float4* lp = (const float4*)logits;
    const int j = nbr[i * DD + lane];     // lane d -> neighbor d (coalesced)
    const int pred = argmax16(lp, j);     // neighbor class pred
    const int cpred = argmax16(lp, i);    // center pred (broadcast load, all lanes)

    // f1 = #neighbors agreeing with center  (one ballot over the full wave)
    wf1 = (float)__popc(ballot32(pred == cpred));

    // f2 = -sum_c p_c * log(p_c), p_c = max(count_c, 1e-5) (clip -> empty bins contribute)
    float f2 = 0.0f;
#pragma unroll
    for (int cls = 0; cls < CC; ++cls) {
      float p = (float)__popc(ballot32(pred == cls));
      p = fmaxf(p, CLAMP_MIN);
      f2 -= p * __logf(p);
    }
    wf2 = f2;

    if (lane == 0) { f1o[i] = wf1; f2o[i] = wf2; }
  }

  // block reduction of (f1, f1^2, f2, f2^2) -> 4 atomics per block
  if (lane == 0) {
    red[wave][0] = wf1; red[wave][1] = wf1 * wf1;
    red[wave][2] = wf2; red[wave][3] = wf2 * wf2;
  }
  __syncthreads();
  if (threadIdx.x < 4) {
    float s = 0.0f;
#pragma unroll
    for (int w = 0; w < WAVES_PER_BLOCK; ++w) s += red[w][threadIdx.x];
    atomicAdd(&stats[threadIdx.x], s);
  }
}

// ---------- kernel 2: finalize LayerNorm scalars ----------
__global__ void k_final(const float* __restrict__ stats, float* __restrict__ scal) {
  if (threadIdx.x == 0) {
    const float invn = 1.0f / (float)NN;
    float m1 = stats[0] * invn;
    float v1 = fmaxf(stats[1] * invn - m1 * m1, 0.0f);   // biased var (jnp.var)
    float m2 = stats[2] * invn;
    float v2 = fmaxf(stats[3] * invn - m2 * m2, 0.0f);
    scal[0] = m1; scal[1] = rsqrtf(v1 + LN_EPS);
    scal[2] = m2; scal[3] = rsqrtf(v2 + LN_EPS);
  }
}

// ---------- kernel 3: gate + gather-sum of neighbor features ----------
__global__ __launch_bounds__(BLOCK) void k_gather(
    const float* __restrict__ h, const int* __restrict__ nbr,
    const float* __restrict__ f1v, const float* __restrict__ f2v,
    const float* __restrict__ old_z, const float* __restrict__ tau1,
    const float* __restrict__ tau2, const float* __restrict__ scal,
    float* __restrict__ new_h, float* __restrict__ z_out) {
  const int wave = threadIdx.x >> 5;
  const int lane = threadIdx.x & 31;
  const int i = blockIdx.x * WAVES_PER_BLOCK + wave;   // uniform per wave
  if (i >= NN) return;

#if __has_builtin(__builtin_amdgcn_global_load_async_to_lds_b32) && \
    __has_builtin(__builtin_amdgcn_s_wait_asynccnt)
  // gfx1250 async-to-LDS path: stage this node's 32 neighbor ids (128B) in LDS
  __shared__ int stage[WAVES_PER_BLOCK][DD];
  typedef __attribute__((address_space(1))) int* gint_t;
  typedef __attribute__((address_space(3))) int* lint_t;
  __builtin_amdgcn_global_load_async_to_lds_b32(
      (gint_t)(nbr + (size_t)i * DD + lane),
      (lint_t)(&stage[wave][lane]), 0, 0);
  __builtin_amdgcn_s_wait_asynccnt(0);
  const int jl = stage[wave][lane];      // lane d holds neighbor id d
#else
  const int jl = nbr[(size_t)i * DD + lane];
#endif

  // Warm L1/WGP$ for this lane's neighbor row while the gate is computed.
  {
    const char* rb = (const char*)(h + (size_t)jl * FF);
    __builtin_prefetch(rb, 0, 1);          // global_prefetch_b8
    __builtin_prefetch(rb + 256, 0, 1);
  }

  // gate / z (uniform across the wave; computed redundantly, broadcast loads)
  const float m1 = scal[0], r1 = scal[1], m2 = scal[2], r2 = scal[3];
  const float a = (f1v[i] - m1) * r1 - tau1[0];
  const float b = (f2v[i] - m2) * r2 - tau2[0];
  // sigmoid(-a) * sigmoid(-b) = 1 / ((1+e^a)(1+e^b))
  const float z = 1.0f / ((1.0f + __expf(a)) * (1.0f + __expf(b)));
  const float gate = fminf(old_z[i], z);
  if (lane == 0) z_out[i] = z;

  // gather-sum: lane l owns float4 column l (32 lanes * 16B = one 512B row)
  const float4* __restrict__ hp = (const float4*)h;
  float4 acc = {0.0f, 0.0f, 0.0f, 0.0f};
#pragma unroll 8
  for (int d = 0; d < DD; ++d) {
    const int jd = bcast_lane(jl, d);                  // uniform row id -> SGPR
    const float4 v = hp[(size_t)jd * (FF / 4) + lane]; // coalesced 512B row
    acc.x += v.x; acc.y += v.y; acc.z += v.z; acc.w += v.w;
  }

  const float4 own = hp[(size_t)i * (FF / 4) + lane];
  float4 o;
  o.x = own.x + gate * fmaxf(acc.x, 0.0f);
  o.y = own.y + gate * fmaxf(acc.y, 0.0f);
  o.z = own.z + gate * fmaxf(acc.z, 0.0f);
  o.w = own.w + gate * fmaxf(acc.w, 0.0f);
  ((float4*)new_h)[(size_t)i * (FF / 4) + lane] = o;
}

extern "C" void kernel_launch(void* const* d_in, const int* in_sizes, int n_in,
                              void* d_out, int out_size, void* d_ws, size_t ws_size,
                              hipStream_t stream) {
  const float* h      = (const float*)d_in[0];
  const float* logits = (const float*)d_in[1];
  const float* old_z  = (const float*)d_in[2];
  const float* tau1   = (const float*)d_in[3];
  const float* tau2   = (const float*)d_in[4];
  const int*   nbr    = (const int*)d_in[5];

  float* new_h = (float*)d_out;                       // [N, F]
  float* z_out = (float*)d_out + (size_t)NN * FF;     // [N]

  float* f1    = (float*)d_ws;          // N
  float* f2    = f1 + NN;               // N
  float* stats = f2 + NN;               // 4: sum1, sumsq1, sum2, sumsq2
  float* scal  = stats + 4;             // 4: mean1, rstd1, mean2, rstd2

  const int blocks = (NN + WAVES_PER_BLOCK - 1) / WAVES_PER_BLOCK;

  k_init<<<1, 32, 0, stream>>>(stats);
  k_stats<<<blocks, BLOCK, 0, stream>>>(logits, nbr, f1, f2, stats);
  k_final<<<1, 32, 0, stream>>>(stats, scal);
  k_gather<<<blocks, BLOCK, 0, stream>>>(h, nbr, f1, f2, old_z, tau1, tau2,
                                         scal, new_h, z_out);
}